// DetectionLayer_22797686407716
// MI455X (gfx1250) — compile-verified
//
#include <hip/hip_runtime.h>
#include <math.h>

// ---------------- problem constants (from reference) ----------------
#define BSZ      128
#define FH       50
#define FW       76
#define FHW      (FH * FW)          // 3800
#define NA       9
#define NANCH    (FHW * NA)         // 34200
#define PRE_NMS  6000
#define KEEP_N   300
#define IOU_THR  0.7f
#define CONF_THR 0.5f
#define NEGF     (-1.0e9f)

#define NMS_T    1024
#define ITEMS    ((PRE_NMS + NMS_T - 1) / NMS_T)   // 6 items per thread

// =====================================================================
// Kernel 3 (placed first so the disasm snippet shows the CDNA5 async
// staging): greedy NMS, one workgroup per image (32 wave32 waves).
// The 120 KB candidate blob (only fits in CDNA5's 320 KB LDS/WGP) is
// staged global->LDS with global_load_async_to_lds_b128 + s_wait_asynccnt,
// then 300 rounds of {2-level shfl_xor argmax, register suppression}.
// =====================================================================
__global__ void __launch_bounds__(NMS_T)
det_nms(const float* __restrict__ blob, float* __restrict__ out) {
    extern __shared__ float lds[];            // 5*PRE_NMS floats = 120000 B
    __shared__ float red_val[32];
    __shared__ int   red_idx[32];
    __shared__ float s_bv;
    __shared__ int   s_bi;

    const int b   = blockIdx.x;
    const int tid = threadIdx.x;
    const float* src = blob + (size_t)b * (5 * PRE_NMS);

    // ---- async bulk copy: 7500 x 16B chunks, 16B per lane per issue ----
    // LDS byte address == low 32 bits of the flat pointer (ISA 10.2).
    {
        unsigned lds_base = (unsigned)(unsigned long long)(void*)lds;
        unsigned long long gbase = (unsigned long long)(const void*)src;
        for (int c = tid; c < (5 * PRE_NMS * 4) / 16; c += NMS_T) {
            unsigned           laddr = lds_base + (unsigned)c * 16u;
            unsigned long long gaddr = gbase + (unsigned long long)c * 16u;
            asm volatile("global_load_async_to_lds_b128 %0, %1, off"
                         :: "v"(laddr), "v"(gaddr) : "memory");
        }
        asm volatile("s_wait_asynccnt 0" ::: "memory");
    }
    __syncthreads();

    // ---- pull this thread's items into registers ----
    float bx1[ITEMS], by1[ITEMS], bx2[ITEMS], by2[ITEMS], ssc[ITEMS], sar[ITEMS];
    for (int k = 0; k < ITEMS; ++k) {
        int i = tid + k * NMS_T;
        if (i < PRE_NMS) {
            bx1[k] = lds[0 * PRE_NMS + i];
            by1[k] = lds[1 * PRE_NMS + i];
            bx2[k] = lds[2 * PRE_NMS + i];
            by2[k] = lds[3 * PRE_NMS + i];
            ssc[k] = lds[4 * PRE_NMS + i];
        } else {
            bx1[k] = by1[k] = bx2[k] = by2[k] = 0.0f;
            ssc[k] = -4.0e9f;                 // below suppressed level
        }
        sar[k] = (bx2[k] - bx1[k]) * (by2[k] - by1[k]);
    }

    float* prop_base  = out + (size_t)b * KEEP_N * 5;
    float* valid_base = out + (size_t)BSZ * KEEP_N * 5 + (size_t)b * KEEP_N;

    for (int it = 0; it < KEEP_N; ++it) {
        // local argmax (tie -> lowest global index by construction)
        float lv = -1.0e30f;
        int   li = 0;
        for (int k = 0; k < ITEMS; ++k) {
            if (ssc[k] > lv) { lv = ssc[k]; li = tid + k * NMS_T; }
        }
        // wave32 reduction
        for (int m = 16; m > 0; m >>= 1) {
            float ov = __shfl_xor(lv, m, 32);
            int   oi = __shfl_xor(li, m, 32);
            if (ov > lv || (ov == lv && oi < li)) { lv = ov; li = oi; }
        }
        if ((tid & 31) == 0) { red_val[tid >> 5] = lv; red_idx[tid >> 5] = li; }
        __syncthreads();
        if (tid < 32) {
            lv = red_val[tid];
            li = red_idx[tid];
            for (int m = 16; m > 0; m >>= 1) {
                float ov = __shfl_xor(lv, m, 32);
                int   oi = __shfl_xor(li, m, 32);
                if (ov > lv || (ov == lv && oi < li)) { lv = ov; li = oi; }
            }
            if (tid == 0) { s_bv = lv; s_bi = li; }
        }
        __syncthreads();

        const float bv = s_bv;
        const int   bi = s_bi;
        const float X1 = lds[0 * PRE_NMS + bi];
        const float Y1 = lds[1 * PRE_NMS + bi];
        const float X2 = lds[2 * PRE_NMS + bi];
        const float Y2 = lds[3 * PRE_NMS + bi];

        if (tid == 0) {
            bool  ok = bv > (0.5f * NEGF);
            float f  = (ok && (bv > CONF_THR)) ? 1.0f : 0.0f;
            float* pr = prop_base + (size_t)it * 5;
            pr[0] = X1 * f; pr[1] = Y1 * f; pr[2] = X2 * f; pr[3] = Y2 * f;
            pr[4] = bv * f;
            valid_base[it] = f;
        }

        // suppress in registers (IoU > thr, plus the picked box itself)
        const float areaB = (X2 - X1) * (Y2 - Y1);
        for (int k = 0; k < ITEMS; ++k) {
            int i = tid + k * NMS_T;
            if (i < PRE_NMS) {
                float ix1 = fmaxf(X1, bx1[k]);
                float iy1 = fmaxf(Y1, by1[k]);
                float ix2 = fminf(X2, bx2[k]);
                float iy2 = fminf(Y2, by2[k]);
                float inter = fmaxf(ix2 - ix1, 0.0f) * fmaxf(iy2 - iy1, 0.0f);
                float iou   = inter / (areaB + sar[k] - inter + 1e-9f);
                if (iou > IOU_THR || i == bi) ssc[k] = NEGF;
            }
        }
        // next round's first barrier provides the needed separation
    }
}

// =====================================================================
// Kernel 1: scores = softmax(p)[...,1] = sigmoid(p1 - p0)
// Thread mapping (b, a, cell) with cell fastest: both logit reads are
// fully coalesced (this is the only HBM-bound stage). Score writes are
// stride-NA but stay within cachelines; L2 (192 MB) absorbs them.
// =====================================================================
__global__ void det_decode_scores(const float* __restrict__ preds,
                                  float* __restrict__ scores) {
    int tid = blockIdx.x * blockDim.x + threadIdx.x;
    if (tid >= BSZ * NA * FHW) return;
    int cell = tid % FHW;
    int t2   = tid / FHW;
    int a    = t2 % NA;
    int b    = t2 / NA;
    const float* base = preds + ((size_t)b * (2 * NA) + 2 * a) * FHW + cell;
    __builtin_prefetch(base + FHW, 0, 1);     // -> global_prefetch_b8 (CDNA5)
    float p0 = base[0];
    float p1 = base[FHW];
    scores[(size_t)b * NANCH + (size_t)cell * NA + a] = 1.0f / (1.0f + __expf(p0 - p1));
}

// =====================================================================
// Kernel 2: per-image exact top-PRE_NMS selection via 4-pass radix
// select on float bits (scores >= 0 so uint order == float order),
// then LDS-atomic compaction + box decode for the 6000 survivors.
// blob layout per image (SoA, 16B-aligned blocks): x1[6000] y1[6000]
// x2[6000] y2[6000] sc[6000]  == 120000 bytes.
// =====================================================================
__global__ void __launch_bounds__(1024)
det_topk(const float* __restrict__ scores,
         const float* __restrict__ regs,
         const float* __restrict__ dboxes,
         const int*   __restrict__ img,
         float* __restrict__ blob) {
    __shared__ unsigned hist[256];
    __shared__ unsigned sh_prefix, sh_remaining, sh_cnt_gt, sh_cnt_eq;

    const int b   = blockIdx.x;
    const int tid = threadIdx.x;
    const float* sc = scores + (size_t)b * NANCH;

    if (tid == 0) { sh_prefix = 0u; sh_remaining = PRE_NMS; }

    for (int r = 0; r < 4; ++r) {
        if (tid < 256) hist[tid] = 0u;
        __syncthreads();
        const unsigned pf    = sh_prefix;
        const int      shift = 24 - 8 * r;
        for (int n = tid; n < NANCH; n += 1024) {
            unsigned u = __float_as_uint(sc[n]);
            bool match = (r == 0) ? true : ((u >> (unsigned)(shift + 8)) == pf);
            if (match) atomicAdd(&hist[(u >> shift) & 0xFFu], 1u);
        }
        __syncthreads();
        if (tid == 0) {
            unsigned rem = sh_remaining;
            unsigned sel = 0u;
            for (int bin = 255; bin >= 0; --bin) {
                unsigned c = hist[bin];
                if (c >= rem) { sel = (unsigned)bin; break; }
                rem -= c;
            }
            sh_prefix    = (sh_prefix << 8) | sel;
            sh_remaining = rem;
        }
        __syncthreads();
    }

    const unsigned T       = sh_prefix;      // exact threshold bits
    const unsigned rem     = sh_remaining;   // how many == T to take
    const unsigned gt_base = PRE_NMS - rem;  // count of strictly-greater
    if (tid == 0) { sh_cnt_gt = 0u; sh_cnt_eq = 0u; }
    __syncthreads();

    const float Hf = (float)img[0];
    const float Wf = (float)img[1];
    const float* regsb = regs + (size_t)b * (4 * NA * FHW);
    float* bb = blob + (size_t)b * (5 * PRE_NMS);

    for (int n = tid; n < NANCH; n += 1024) {
        unsigned u = __float_as_uint(sc[n]);
        int slot = -1;
        if (u > T) {
            slot = (int)atomicAdd(&sh_cnt_gt, 1u);
        } else if (u == T) {
            unsigned e = atomicAdd(&sh_cnt_eq, 1u);
            if (e < rem) slot = (int)(gt_base + e);
        }
        if (slot >= 0) {
            int a    = n % NA;
            int cell = n / NA;
            float d0 = dboxes[4 * n + 0], d1 = dboxes[4 * n + 1];
            float d2 = dboxes[4 * n + 2], d3 = dboxes[4 * n + 3];
            float w  = d2 - d0, h = d3 - d1;
            float cx = d0 + 0.5f * w, cy = d1 + 0.5f * h;
            size_t ro = (size_t)(4 * a) * FHW + cell;
            float r0 = regsb[ro];
            float r1 = regsb[ro + 1 * FHW];
            float r2 = regsb[ro + 2 * FHW];
            float r3 = regsb[ro + 3 * FHW];
            float pcx = r0 * w + cx, pcy = r1 * h + cy;
            float pw = __expf(r2) * w, ph = __expf(r3) * h;
            float x1 = fminf(fmaxf(pcx - 0.5f * pw, 0.0f), Wf);
            float y1 = fminf(fmaxf(pcy - 0.5f * ph, 0.0f), Hf);
            float x2 = fminf(fmaxf(pcx + 0.5f * pw, 0.0f), Wf);
            float y2 = fminf(fmaxf(pcy + 0.5f * ph, 0.0f), Hf);
            bb[0 * PRE_NMS + slot] = x1;
            bb[1 * PRE_NMS + slot] = y1;
            bb[2 * PRE_NMS + slot] = x2;
            bb[3 * PRE_NMS + slot] = y2;
            bb[4 * PRE_NMS + slot] = __uint_as_float(u);
        }
    }
}

// =====================================================================
// launcher
// =====================================================================
extern "C" void kernel_launch(void* const* d_in, const int* in_sizes, int n_in,
                              void* d_out, int out_size, void* d_ws, size_t ws_size,
                              hipStream_t stream) {
    (void)in_sizes; (void)n_in; (void)out_size; (void)ws_size;
    const float* preds = (const float*)d_in[0];
    const float* regs  = (const float*)d_in[1];
    const int*   img   = (const int*)d_in[2];
    const float* dbox  = (const float*)d_in[3];
    float* out = (float*)d_out;

    float* scores_ws = (float*)d_ws;                        // BSZ*NANCH floats
    float* blob_ws   = scores_ws + (size_t)BSZ * NANCH;     // BSZ*5*PRE_NMS floats

    const int total = BSZ * NA * FHW;
    det_decode_scores<<<(total + 255) / 256, 256, 0, stream>>>(preds, scores_ws);
    det_topk<<<BSZ, 1024, 0, stream>>>(scores_ws, regs, dbox, img, blob_ws);
    det_nms<<<BSZ, NMS_T, 5 * PRE_NMS * sizeof(float), stream>>>(blob_ws, out);
}